// DyRep_5875515261375
// MI455X (gfx1250) — compile-verified
//
#include <hip/hip_runtime.h>
#include <hip/hip_bf16.h>
#include <math.h>

#define N_NODES 4096
#define D 128
#define B_EV 128
#define Q 5
#define MAXNB 128          // neighbor capacity per agg (mean ~32, P(>128) ~ 0)
#define NT 1024            // 32 wave32s, one WGP
#define ZPAD 132           // LDS row pitch (floats) -> conflict-free strided reads

typedef float v2f __attribute__((ext_vector_type(2)));
typedef float v8f __attribute__((ext_vector_type(8)));

__device__ __forceinline__ float sigm(float x) { return 1.0f / (1.0f + expf(-x)); }
// numerically stable softplus, matches jax.nn.softplus
__device__ __forceinline__ float splus(float x) {
    return fmaxf(x, 0.0f) + log1pf(expf(-fabsf(x)));
}

__global__ void dyrep_copy_f32(const float* __restrict__ src, float* __restrict__ dst, int n) {
    int i = blockIdx.x * blockDim.x + threadIdx.x;
    if (i < n) dst[i] = src[i];
}

__global__ __launch_bounds__(NT, 1)
void dyrep_scan(const int* __restrict__ u, const int* __restrict__ v,
                const int* __restrict__ et, const float* __restrict__ time_diff,
                const int* __restrict__ neg, float* __restrict__ z,
                const float* __restrict__ A, const float* __restrict__ S,
                const float* __restrict__ w_om0, const float* __restrict__ b_om0,
                const float* __restrict__ w_om1, const float* __restrict__ b_om1,
                const float* __restrict__ psi,
                const float* __restrict__ W_h, const float* __restrict__ b_h,
                const float* __restrict__ W_struct, const float* __restrict__ b_struct,
                const float* __restrict__ W_rec, const float* __restrict__ b_rec,
                const float* __restrict__ W_t, const float* __restrict__ b_t,
                float* __restrict__ out)
{
    extern __shared__ char smem[];
    float* sWhT = (float*)smem;          // [128][128]  WhT[k][n] = W_h[n][k]
    float* sZg  = sWhT + D * D;          // [MAXNB][ZPAD] gathered neighbor z rows

    __shared__ float sQ[MAXNB];
    __shared__ int   sNb[MAXNB];
    __shared__ int   sHstrI[2][D];       // float bits; atomicMax on nonneg floats
    __shared__ float sZu[D], sZv[D];
    __shared__ float sZupd[2][D];
    __shared__ float sWom[2][2 * D];
    __shared__ float sBh[D];
    __shared__ float sTd[2][4];
    __shared__ int   sCnt;
    __shared__ float sSum;

    const int tid  = threadIdx.x;
    const int lane = tid & 31;
    const int wave = tid >> 5;

    // one-time preload: W_h transposed into LDS, omega vectors, b_h
    for (int i = tid; i < D * D; i += NT) {
        int n = i >> 7, k = i & 127;
        sWhT[k * D + n] = W_h[i];
    }
    for (int i = tid; i < 2 * D; i += NT) { sWom[0][i] = w_om0[i]; sWom[1][i] = w_om1[i]; }
    if (tid < D) sBh[tid] = b_h[tid];
    __syncthreads();

    const float psi0 = psi[0], psi1 = psi[1];
    const float bo0 = b_om0[0], bo1 = b_om1[0];

    for (int step = 0; step < B_EV; ++step) {
        const int ui = u[step], vi = v[step];

        if (tid < D)            sZu[tid]       = z[(size_t)ui * D + tid];
        else if (tid < 2 * D)   sZv[tid - D]   = z[(size_t)vi * D + (tid - D)];
        else if (tid < 2 * D + 8) {
            int k = (tid - 2 * D) >> 2, c = (tid - 2 * D) & 3;
            const float tsd[4] = {50.f, 7.f, 15.f, 15.f};
            sTd[k][c] = time_diff[step * 8 + k * 4 + c] / tsd[c];
        }
        __syncthreads();

        // ---- intensity outputs (use pre-update z): waves 0..10, one per output column ----
        if (wave < 11) {
            const float* xu; const float* xv;
            if (wave == 0) { xu = sZu; xv = sZv; }
            else {
                int kk = wave - 1;
                int nidx = neg[step * (2 * Q) + kk];
                if (kk < Q) { xu = sZu;                     xv = z + (size_t)nidx * D; }
                else        { xu = z + (size_t)nidx * D;    xv = sZv; }
            }
            float g0 = 0.f, g1 = 0.f;
            for (int j = lane; j < D; j += 32) {
                float a = xu[j], b = xv[j];
                g0 += a * sWom[0][j] + b * sWom[0][D + j];
                g1 += a * sWom[1][j] + b * sWom[1][D + j];
            }
            for (int off = 16; off > 0; off >>= 1) {
                g0 += __shfl_xor(g0, off, 32);
                g1 += __shfl_xor(g1, off, 32);
            }
            if (lane == 0) {
                g0 += bo0; g1 += bo1;
                float lam;
                if (wave == 0) {
                    int e   = et[step];
                    float p = e ? psi1 : psi0;
                    float g = e ? g1 : g0;
                    lam = p * splus(g / (p + 1e-7f));
                } else {
                    lam = psi0 * splus(g0 / (psi0 + 1e-7f)) +
                          psi1 * splus(g1 / (psi1 + 1e-7f));
                }
                out[step * 11 + wave] = lam;
            }
        }

        // ---- two aggregation passes: h_struct[0]=agg(vi), h_struct[1]=agg(ui) ----
        for (int pass = 0; pass < 2; ++pass) {
            const int other = (pass == 0) ? vi : ui;
            if (tid == 0) { sCnt = 0; sSum = 0.f; }
            if (tid < D)  sHstrI[pass][tid] = 0;     // 0.0f bits
            __syncthreads();

            // scan A[other] (4096 floats = 1024 threads x float4); build neighbor list
            {
                const float4 a4 = ((const float4*)(A + (size_t)other * N_NODES))[tid];
                const float av[4] = {a4.x, a4.y, a4.z, a4.w};
                #pragma unroll
                for (int c = 0; c < 4; ++c) {
                    if (av[c] > 0.f) {
                        int pos = atomicAdd(&sCnt, 1);
                        if (pos < MAXNB) {
                            float e = expf(S[(size_t)other * N_NODES + tid * 4 + c]);
                            sNb[pos] = tid * 4 + c;
                            sQ[pos]  = e;
                            atomicAdd(&sSum, e);
                        }
                    }
                }
            }
            __syncthreads();
            const int cnt    = min(sCnt, MAXNB);
            const int mtiles = (cnt + 15) >> 4;
            const int Mpad   = mtiles << 4;
            const float qscale = 1.f / (sSum + 1e-7f);
            if (tid < Mpad) sQ[tid] = (tid < cnt) ? sQ[tid] * qscale : 0.f;

            // gather neighbor z rows to LDS (float4 chunks), zero pad rows
            for (int i = tid; i < (Mpad << 5); i += NT) {
                int j = i >> 5, c = i & 31;
                float4 val = make_float4(0.f, 0.f, 0.f, 0.f);
                if (j < cnt) val = ((const float4*)(z + (size_t)sNb[j] * D))[c];
                *(float4*)(sZg + j * ZPAD + c * 4) = val;
            }
            __syncthreads();

            // fp32 WMMA GEMM: Whz[m][d] = sum_k Znb[m][k] * W_h[d][k]; then sigmoid+max
            const int tiles = mtiles * 8;      // (mtile, ntile) pairs, ntiles = 128/16 = 8
            const int half  = lane >> 4;       // 0: lanes 0-15, 1: lanes 16-31
            const int lm    = lane & 15;
            for (int t = wave; t < tiles; t += 32) {
                const int mt = t >> 3, nt = t & 7;
                const float* arow = sZg + (mt * 16 + lm) * ZPAD;   // M = mt*16+lm
                const int dcol = nt * 16 + lm;                     // N = nt*16+lm
                v8f acc = {0.f, 0.f, 0.f, 0.f, 0.f, 0.f, 0.f, 0.f};
                for (int k0 = 0; k0 < D; k0 += 4) {
                    const int ka = k0 + half * 2;                  // K per ISA A/B layout
                    v2f a; a.x = arow[ka];                 a.y = arow[ka + 1];
                    v2f b; b.x = sWhT[ka * D + dcol];      b.y = sWhT[(ka + 1) * D + dcol];
                    acc = __builtin_amdgcn_wmma_f32_16x16x4_f32(
                              false, a, false, b, (short)0, acc, false, false);
                }
                const float bh = sBh[dcol];
                #pragma unroll
                for (int r = 0; r < 8; ++r) {
                    int m = mt * 16 + r + 8 * half;                // C layout: VGPR r -> M=r / r+8
                    if (m < cnt) {
                        float val = sigm(sQ[m] * (acc[r] + bh));
                        atomicMax(&sHstrI[pass][dcol], __float_as_int(val));
                    }
                }
            }
            __syncthreads();
        }

        // ---- z_upd = sigmoid(W_struct h + W_rec z_pair + W_t td + biases) ----
        if (tid < 2 * D) {
            const int k = tid >> 7, d = tid & 127;
            float s = b_struct[d] + b_rec[d] + b_t[d];
            const float* ws = W_struct + d * D;
            const float* wr = W_rec + d * D;
            const float* zp = k ? sZv : sZu;
            const int*   hI = sHstrI[k];
            float acc = 0.f;
            for (int e = 0; e < D; ++e)
                acc += ws[e] * __int_as_float(hI[e]) + wr[e] * zp[e];
            s += acc;
            const float* wt = W_t + d * 4;
            s += wt[0] * sTd[k][0] + wt[1] * sTd[k][1] + wt[2] * sTd[k][2] + wt[3] * sTd[k][3];
            sZupd[k][d] = sigm(s);
        }
        __syncthreads();
        // reference semantics: write ui first, then vi (vi wins if ui==vi)
        if (tid < D) z[(size_t)ui * D + tid] = sZupd[0][tid];
        __syncthreads();
        if (tid < D) z[(size_t)vi * D + tid] = sZupd[1][tid];
        __syncthreads();
    }
}

extern "C" void kernel_launch(void* const* d_in, const int* in_sizes, int n_in,
                              void* d_out, int out_size, void* d_ws, size_t ws_size,
                              hipStream_t stream) {
    const int*   u         = (const int*)d_in[0];
    const int*   v         = (const int*)d_in[1];
    const int*   et        = (const int*)d_in[2];
    const float* time_diff = (const float*)d_in[3];
    const int*   neg       = (const int*)d_in[4];
    const float* z0        = (const float*)d_in[5];
    const float* A         = (const float*)d_in[6];
    const float* S         = (const float*)d_in[7];
    const float* w_om0     = (const float*)d_in[8];
    const float* b_om0     = (const float*)d_in[9];
    const float* w_om1     = (const float*)d_in[10];
    const float* b_om1     = (const float*)d_in[11];
    const float* psi       = (const float*)d_in[12];
    const float* W_h       = (const float*)d_in[13];
    const float* b_h       = (const float*)d_in[14];
    const float* W_struct  = (const float*)d_in[15];
    const float* b_struct  = (const float*)d_in[16];
    const float* W_rec     = (const float*)d_in[17];
    const float* b_rec     = (const float*)d_in[18];
    const float* W_t       = (const float*)d_in[19];
    const float* b_t       = (const float*)d_in[20];

    float* z   = (float*)d_ws;           // mutable copy of z0 (2 MB)
    float* out = (float*)d_out;

    const int zn = N_NODES * D;
    dyrep_copy_f32<<<(zn + 255) / 256, 256, 0, stream>>>(z0, z, zn);

    const size_t smem = (size_t)(D * D + MAXNB * ZPAD) * sizeof(float); // 130 KB
    dyrep_scan<<<1, NT, smem, stream>>>(u, v, et, time_diff, neg, z, A, S,
                                        w_om0, b_om0, w_om1, b_om1, psi,
                                        W_h, b_h, W_struct, b_struct,
                                        W_rec, b_rec, W_t, b_t, out);
}